// GCN_84937273246041
// MI455X (gfx1250) — compile-verified
//
#include <hip/hip_runtime.h>

// ---------------------------------------------------------------------------
// GCN forward: out = (A_hat @ relu(A_hat @ X @ W1^T) * mask) @ W2^T
// MI455X / gfx1250: wave32, fp32 WMMA (v_wmma_f32_16x16x4_f32) for the dense
// layers (fp32 kept for validation accuracy; the GEMMs are compute-trivial),
// L2-resident COO scatter-add SpMM via hardware global_atomic_add_f32.
//
// GEMM: one wave owns a 32x32 output tile (2x2 WMMA tiles) -> 1 b64 load per
// WMMA instead of 2, 4x register reuse of A/B fragments.
// SpMM: one wave per edge, b128 gather per lane, wave-uniform index loads.
// ---------------------------------------------------------------------------

typedef __attribute__((ext_vector_type(2))) float v2f;
typedef __attribute__((ext_vector_type(8))) float v8f;

#define NNODES 100000
#define INDIM  128
#define HIDDIM 128
#define OUTDIM 64

// ----------------------------- SpMM (COO scatter-add) ----------------------
// out[row[e], :] += vals[e] * X[col[e], :].  One wave per edge; lane l covers
// features [4l, 4l+4) with a single b128 gather and 4 fp32 hardware atomics.
// row/col/vals loads are wave-uniform (single cacheline request); the 512B
// row gather is fully coalesced; atomics land in the 192MB L2 (the whole
// 51MB feature matrix is L2-resident).
template <int D>
__global__ void gcn_spmm_scatter(const int* __restrict__ row,
                                 const int* __restrict__ col,
                                 const float* __restrict__ vals,
                                 const float* __restrict__ X,
                                 float* __restrict__ Y,
                                 int E) {
  static_assert(D == 32 * 4, "one wave covers D floats as float4 per lane");
  const int lane = threadIdx.x & 31;
  const int wavesPerBlock = blockDim.x >> 5;
  const int totalWaves = gridDim.x * wavesPerBlock;
  int e = blockIdx.x * wavesPerBlock + (threadIdx.x >> 5);
  for (; e < E; e += totalWaves) {
    const int r = row[e];
    const int c = col[e];
    const float v = vals[e];
    const float4 x = *((const float4*)(X + (long long)c * D) + lane);
    float* yp = Y + (long long)r * D + (lane << 2);
    unsafeAtomicAdd(yp + 0, v * x.x);
    unsafeAtomicAdd(yp + 1, v * x.y);
    unsafeAtomicAdd(yp + 2, v * x.z);
    unsafeAtomicAdd(yp + 3, v * x.w);
  }
}

// ----------------------------- Dense GEMM via WMMA -------------------------
// Dst[m,n] = sum_k A[m,k] * W[n,k]   (A @ W^T), optionally fused relu*mask.
// One wave computes an (MT*16) x (NT*16) tile: per k-step it loads MT A-frags
// + NT B-frags (b64 each) and issues MT*NT v_wmma_f32_16x16x4_f32, fully
// unrolled over K.
//
// Fragment layouts (CDNA5 ISA 7.12.2, 32-bit, wave32):
//   A 16x4 : lanes 0-15 -> M=lane, K={0,1} in VGPR0/1; lanes 16-31 -> K={2,3}
//   B 4x16 : lanes 0-15 -> N=lane, K={0,1};            lanes 16-31 -> K={2,3}
//   C/D    : 8 VGPRs; VGPR r holds M = r + 8*(lane>=16), N = lane & 15
template <int K, int NCOLS, int MT, int NT, bool FUSE_RELU_MASK>
__global__ void gcn_gemm_wmma_f32(const float* __restrict__ A,
                                  const float* __restrict__ W,
                                  float* __restrict__ Dst,
                                  const float* __restrict__ mask) {
  const int lane  = threadIdx.x & 31;
  const int lrow  = lane & 15;        // M row (A) / N col (B) within a tile
  const int khalf = (lane >> 4) << 1; // 0 for lanes 0-15, 2 for lanes 16-31

  const int mrow = blockIdx.x * (16 * MT);  // node tile base
  const int ncol = blockIdx.y * (16 * NT);  // output-column tile base

  const float* ap[MT];
#pragma unroll
  for (int mt = 0; mt < MT; ++mt)
    ap[mt] = A + (long long)(mrow + mt * 16 + lrow) * K + khalf;
  const float* bp[NT];
#pragma unroll
  for (int nt = 0; nt < NT; ++nt)
    bp[nt] = W + (long long)(ncol + nt * 16 + lrow) * K + khalf;

  v8f acc[MT][NT] = {};
#pragma unroll
  for (int k = 0; k < K; k += 4) {
    v2f a[MT], b[NT];
#pragma unroll
    for (int mt = 0; mt < MT; ++mt) {
      a[mt].x = ap[mt][k];
      a[mt].y = ap[mt][k + 1];
    }
#pragma unroll
    for (int nt = 0; nt < NT; ++nt) {
      b[nt].x = bp[nt][k];
      b[nt].y = bp[nt][k + 1];
    }
#pragma unroll
    for (int mt = 0; mt < MT; ++mt)
#pragma unroll
      for (int nt = 0; nt < NT; ++nt)
        // 8 args: (neg_a, A, neg_b, B, c_mod, C, reuse_a, reuse_b)
        acc[mt][nt] = __builtin_amdgcn_wmma_f32_16x16x4_f32(
            false, a[mt], false, b[nt], (short)0, acc[mt][nt], false, false);
  }

  const int hi = lane >> 4;  // 0: rows r..; 1: rows r+8
#pragma unroll
  for (int mt = 0; mt < MT; ++mt) {
#pragma unroll
    for (int nt = 0; nt < NT; ++nt) {
      const int n = ncol + nt * 16 + lrow;
      const int mbase = mrow + mt * 16 + (hi << 3);
#pragma unroll
      for (int r = 0; r < 8; ++r) {
        const int m = mbase + r;
        float v = acc[mt][nt][r];
        if (FUSE_RELU_MASK) {
          v = fmaxf(v, 0.0f) * mask[(long long)m * NCOLS + n];
        }
        Dst[(long long)m * NCOLS + n] = v;
      }
    }
  }
}

// ---------------------------------------------------------------------------
extern "C" void kernel_launch(void* const* d_in, const int* in_sizes, int n_in,
                              void* d_out, int out_size, void* d_ws, size_t ws_size,
                              hipStream_t stream) {
  // setup_inputs() order: X, row, col, vals, W1, W2, drop_mask
  const float* X    = (const float*)d_in[0];
  const int*   row  = (const int*)  d_in[1];
  const int*   col  = (const int*)  d_in[2];
  const float* vals = (const float*)d_in[3];
  const float* W1   = (const float*)d_in[4];
  const float* W2   = (const float*)d_in[5];
  const float* mask = (const float*)d_in[6];
  const int E = in_sizes[1];

  float* H1 = (float*)d_ws;                       // [N, 128] spmm(X)
  float* H2 = H1 + (size_t)NNODES * INDIM;        // [N, 128] relu(H1@W1^T)*mask
  float* H3 = H2 + (size_t)NNODES * HIDDIM;       // [N, 128] spmm(H2)
  float* out = (float*)d_out;                     // [N, 64]

  const int threads = 256;                        // 8 waves per block
  const int wavesPerBlock = threads / 32;
  long long eBlocks = ((long long)E + wavesPerBlock - 1) / wavesPerBlock;
  if (eBlocks > (1ll << 20)) eBlocks = (1ll << 20);  // grid-stride cap

  // ---- layer 1: H1 = A_hat @ X -------------------------------------------
  hipMemsetAsync(H1, 0, (size_t)NNODES * INDIM * sizeof(float), stream);
  gcn_spmm_scatter<INDIM><<<(int)eBlocks, threads, 0, stream>>>(
      row, col, vals, X, H1, E);

  // ---- H2 = relu(H1 @ W1^T) * mask ---------------------------------------
  {
    dim3 grid(NNODES / 32, HIDDIM / 32);
    gcn_gemm_wmma_f32<INDIM, HIDDIM, 2, 2, true>
        <<<grid, 32, 0, stream>>>(H1, W1, H2, mask);
  }

  // ---- layer 2: H3 = A_hat @ H2 ------------------------------------------
  hipMemsetAsync(H3, 0, (size_t)NNODES * HIDDIM * sizeof(float), stream);
  gcn_spmm_scatter<HIDDIM><<<(int)eBlocks, threads, 0, stream>>>(
      row, col, vals, H2, H3, E);

  // ---- out = H3 @ W2^T ----------------------------------------------------
  {
    dim3 grid(NNODES / 32, OUTDIM / 32);
    gcn_gemm_wmma_f32<HIDDIM, OUTDIM, 2, 2, false>
        <<<grid, 32, 0, stream>>>(H3, W2, out, nullptr);
  }
}